// SEM_40200893890649
// MI455X (gfx1250) — compile-verified
//
#include <hip/hip_runtime.h>
#include <hip/hip_bf16.h>
#include <math.h>

// Problem constants (match reference)
#define BB 32
#define NN 2000
#define DD 128
#define RNN 640
#define KTOP 64
#define D2 16384   // DD*DD

typedef float v2f __attribute__((ext_vector_type(2)));
typedef float v8f __attribute__((ext_vector_type(8)));

// ---------------------------------------------------------------------------
// Phase 1a: scorer[b,d] = tanh(ht[b]·mapper_w[d] + mapper_b[d]); invnorm[b]
// ---------------------------------------------------------------------------
__global__ void k_scorer(const float* __restrict__ ht,
                         const float* __restrict__ mw,
                         const float* __restrict__ mb,
                         float* __restrict__ scorer_out,
                         float* __restrict__ invnorm) {
  int b = blockIdx.x;
  int d = threadIdx.x;             // blockDim = 128
  const float* h = ht + (long)b * RNN;
  const float* w = mw + (long)d * RNN;
  float acc = 0.f;
  for (int r = 0; r < RNN; ++r) acc += h[r] * w[r];
  float s = tanhf(acc + mb[d]);
  scorer_out[b * DD + d] = s;
  __shared__ float red[DD];
  red[d] = s * s;
  __syncthreads();
  for (int o = 64; o > 0; o >>= 1) {
    if (d < o) red[d] += red[d + o];
    __syncthreads();
  }
  if (d == 0) invnorm[b] = rsqrtf(red[0]);
}

// ---------------------------------------------------------------------------
// Phase 1b: scores[b,n] = (node_embs[b,n,:]·scorer[b,:]) * invnorm[b]
// one wave per row n; grid (N/8, B), block 256 (8 waves)
// ---------------------------------------------------------------------------
__global__ void k_scores(const float* __restrict__ ne,
                         const float* __restrict__ scorer,
                         const float* __restrict__ invnorm,
                         float* __restrict__ scores) {
  int b = blockIdx.y;
  int wave = threadIdx.x >> 5;
  int lane = threadIdx.x & 31;
  int n = blockIdx.x * 8 + wave;
  const float* row = ne + ((long)b * NN + n) * DD;
  const float* sc = scorer + (long)b * DD;
  float acc = 0.f;
  for (int i = lane; i < DD; i += 32) acc += row[i] * sc[i];
  for (int o = 16; o > 0; o >>= 1) acc += __shfl_down(acc, o, 32);
  if (lane == 0) scores[(long)b * NN + n] = acc * invnorm[b];
}

// ---------------------------------------------------------------------------
// Phase 2: per-batch top-64 (value desc, index asc tiebreak), entropy,
// policy_score. One block of 256 per batch.
// ---------------------------------------------------------------------------
__global__ void k_topk(const float* __restrict__ scores,
                       int* __restrict__ topk,
                       float* __restrict__ entropy_out,
                       float* __restrict__ policy_out) {
  __shared__ float s[NN];
  __shared__ float rv[256];
  __shared__ int ri[256];
  __shared__ float chosenv[KTOP];
  int b = blockIdx.x, t = threadIdx.x;
  const float* sb = scores + (long)b * NN;
  for (int i = t; i < NN; i += 256) s[i] = sb[i];
  __syncthreads();
  // max
  float m = -3.0e38f;
  for (int i = t; i < NN; i += 256) m = fmaxf(m, s[i]);
  rv[t] = m;
  __syncthreads();
  for (int o = 128; o > 0; o >>= 1) {
    if (t < o) rv[t] = fmaxf(rv[t], rv[t + o]);
    __syncthreads();
  }
  float M = rv[0];
  __syncthreads();
  // sum(exp) and sum(exp*(x-M))
  float se = 0.f, te = 0.f;
  for (int i = t; i < NN; i += 256) {
    float e = __expf(s[i] - M);
    se += e;
    te += e * (s[i] - M);
  }
  rv[t] = se;
  __syncthreads();
  for (int o = 128; o > 0; o >>= 1) {
    if (t < o) rv[t] += rv[t + o];
    __syncthreads();
  }
  float S = rv[0];
  __syncthreads();
  rv[t] = te;
  __syncthreads();
  for (int o = 128; o > 0; o >>= 1) {
    if (t < o) rv[t] += rv[t + o];
    __syncthreads();
  }
  float T = rv[0];
  __syncthreads();
  float logZ = logf(S);
  if (t == 0) entropy_out[b] = logZ - T / S;
  // iterative argmax top-64
  for (int sel = 0; sel < KTOP; ++sel) {
    float bv = -3.0e38f;
    int bi = 0x7fffffff;
    for (int i = t; i < NN; i += 256) {
      float v = s[i];
      if (v > bv || (v == bv && i < bi)) { bv = v; bi = i; }
    }
    rv[t] = bv; ri[t] = bi;
    __syncthreads();
    for (int o = 128; o > 0; o >>= 1) {
      if (t < o) {
        if (rv[t + o] > rv[t] || (rv[t + o] == rv[t] && ri[t + o] < ri[t])) {
          rv[t] = rv[t + o]; ri[t] = ri[t + o];
        }
      }
      __syncthreads();
    }
    if (t == 0) {
      chosenv[sel] = rv[0];
      topk[b * KTOP + sel] = ri[0];
      s[ri[0]] = -3.0e38f;   // mask out
    }
    __syncthreads();
  }
  // policy_score = mean over (tiled) idx of logp == mean over topk of logp
  rv[t] = (t < KTOP) ? (chosenv[t] - M - logZ) : 0.f;
  __syncthreads();
  for (int o = 128; o > 0; o >>= 1) {
    if (t < o) rv[t] += rv[t + o];
    __syncthreads();
  }
  if (t == 0) policy_out[b] = rv[0] / (float)KTOP;
}

// ---------------------------------------------------------------------------
// Phase 3a: di[b,j] = (sum_i Ahat[b, idx_i, idx_j])^-1/2   (idx[k]=topk[k%64])
// ---------------------------------------------------------------------------
__global__ void k_di(const float* __restrict__ Ahat,
                     const int* __restrict__ topk,
                     float* __restrict__ di) {
  int b = blockIdx.x, j = threadIdx.x;   // blockDim = 128
  __shared__ int idx[DD];
  idx[j] = topk[b * KTOP + (j & 63)];
  __syncthreads();
  const float* Ab = Ahat + (long)b * NN * NN;
  int cj = idx[j];
  float sum = 0.f;
  for (int i = 0; i < DD; ++i) sum += Ab[(long)idx[i] * NN + cj];
  di[b * DD + j] = rsqrtf(sum);
}

// Phase 3b: Anorm[b,i,j] = di[i]*di[j]*Ahat[b, idx_i, idx_j]
__global__ void k_An(const float* __restrict__ Ahat,
                     const int* __restrict__ topk,
                     const float* __restrict__ di,
                     float* __restrict__ An) {
  int b = blockIdx.x, t = threadIdx.x;   // blockDim = 256
  __shared__ int idx[DD];
  __shared__ float d[DD];
  if (t < DD) { idx[t] = topk[b * KTOP + (t & 63)]; d[t] = di[b * DD + t]; }
  __syncthreads();
  const float* Ab = Ahat + (long)b * NN * NN;
  float* Ao = An + (long)b * D2;
  for (int q = t; q < D2; q += 256) {
    int i = q >> 7, j = q & 127;
    Ao[q] = d[i] * d[j] * Ab[(long)idx[i] * NN + idx[j]];
  }
}

// Phase 3c: ne[b,k,f] = node_embs[b, idx_k, f]; x[b,f,k] = ne * tanh(score[idx_k])
__global__ void k_sel(const float* __restrict__ node_embs,
                      const float* __restrict__ scores,
                      const int* __restrict__ topk,
                      float* __restrict__ x,
                      float* __restrict__ ne) {
  int b = blockIdx.x, t = threadIdx.x;   // blockDim = 256
  __shared__ int idx[DD];
  __shared__ float tsc[DD];
  if (t < DD) {
    int id = topk[b * KTOP + (t & 63)];
    idx[t] = id;
    tsc[t] = tanhf(scores[(long)b * NN + id]);
  }
  __syncthreads();
  const float* nb = node_embs + (long)b * NN * DD;
  float* xb = x + (long)b * D2;
  float* neb = ne + (long)b * D2;
  for (int q = t; q < D2; q += 256) {
    int k = q >> 7, f = q & 127;
    float v = nb[(long)idx[k] * DD + f];
    neb[q] = v;
    xb[f * DD + k] = v * tsc[k];
  }
}

// ---------------------------------------------------------------------------
// Phase 4: generic batched 128x128x128 f32 GEMM via V_WMMA_F32_16X16X4_F32.
// C[b] = epilogue( A1[b]@B1[b] (+ A2[b]@B2[b]) (+ bias) ).
// Batch strides sX == 0 mean the operand is shared across batch.
// Block = 256 (8 waves, one batch per block); wave w owns rows [16w,16w+16),
// 8 column tiles of 16 -> 8 v8f accumulators. K staged in 32-wide LDS panels.
//
// A fragment (16x4 f32): lanes 0-15 = rows M with K0/K1; lanes 16-31 = K2/K3.
//   As is row-major with pitch 36 (bank-conflict-free), K-pairs contiguous ->
//   single aligned v2f (b64) LDS load per fragment.
// B fragment (4x16 f32): V0 = K0 (lanes 0-15) / K2 (16-31), V1 = K1/K3.
//   Bs is stored K-PAIR-INTERLEAVED: element (k,n) at ((k>>1)*128+n)*2+(k&1),
//   so {B[kb][n], B[kb+1][n]} is one aligned v2f (b64) LDS load -> no repack
//   movs before the WMMA, and adjacent nt tiles fuse into ds_load_2addr_b64.
// C tile: VGPR r, lane l -> M = r + 8*(l>=16), N = l&15.
// ---------------------------------------------------------------------------
#define EP_SIGMOID 0
#define EP_SIG_MUL 1   // sigmoid(acc+b) * E1      (rh = reset * prev_D)
#define EP_GRUMIX  2   // (1-E1)*E2 + E1*tanh(acc+b)
#define EP_NONE    3
#define EP_RELU    4
#define EP_OUTMIX  5   // 0.5*(E1 + relu(acc))

__global__ void __launch_bounds__(256) k_gemm128(
    const float* __restrict__ A1, long sA1,
    const float* __restrict__ B1, long sB1,
    const float* __restrict__ A2, long sA2,
    const float* __restrict__ B2, long sB2,
    const float* __restrict__ bias,
    const float* __restrict__ E1,
    const float* __restrict__ E2,
    float* __restrict__ C, long sC, int mode) {
  constexpr int PA = 36;                   // padded A pitch (bank-conflict free)
  __shared__ float As[DD * PA];            // 18 KB
  __shared__ float Bs[32 * DD];            // 16 KB, K-pair interleaved
  int b = blockIdx.x;
  int t = threadIdx.x;
  int wave = t >> 5, lane = t & 31;
  int m0 = wave * 16;
  int lhalf = lane >> 4;                   // 0: K0/K1 half, 1: K2/K3 half
  int l15 = lane & 15;

  v8f acc[8];
#pragma unroll
  for (int i = 0; i < 8; ++i)
#pragma unroll
    for (int r = 0; r < 8; ++r) acc[i][r] = 0.f;

  for (int pass = 0; pass < 2; ++pass) {
    if (pass && A2 == nullptr) break;
    const float* Ab = (pass ? A2 + sA2 * b : A1 + sA1 * b);
    const float* Bb = (pass ? B2 + sB2 * b : B1 + sB1 * b);
    for (int kp = 0; kp < DD; kp += 32) {
      __syncthreads();
      // stage A panel: 128 rows x 32 cols, row-major, pitch PA
#pragma unroll
      for (int i = 0; i < 4; ++i) {
        int q = t + i * 256;
        int row = q >> 3, c4 = (q & 7) * 4;
        float4 v = *(const float4*)(Ab + (long)row * DD + kp + c4);
        *(float4*)(&As[row * PA + c4]) = v;
      }
      // stage B panel: 32 rows x 128 cols, K-pair interleaved:
      // quad {B[k][n], B[k+1][n], B[k][n+1], B[k+1][n+1]} -> one float4 store
#pragma unroll
      for (int i = 0; i < 4; ++i) {
        int q = t + i * 256;                 // 0..1023
        int pr = q >> 6;                     // pair-row 0..15
        int n = (q & 63) * 2;                // 0,2,..,126
        const float* r0 = Bb + (long)(kp + pr * 2) * DD + n;
        const float* r1 = r0 + DD;
        float2 a0 = *(const float2*)r0;
        float2 a1 = *(const float2*)r1;
        float4 w;
        w.x = a0.x; w.y = a1.x; w.z = a0.y; w.w = a1.y;
        *(float4*)(&Bs[(pr * DD + n) * 2]) = w;
      }
      __syncthreads();
      const v2f* Bp = (const v2f*)Bs;
#pragma unroll
      for (int kt = 0; kt < 8; ++kt) {
        int kb = kt * 4 + lhalf * 2;         // even
        v2f a = *(const v2f*)(&As[(m0 + l15) * PA + kb]);
        int pb = (kt * 2 + lhalf) * DD + l15;
#pragma unroll
        for (int nt = 0; nt < 8; ++nt) {
          v2f bf = Bp[pb + nt * 16];
          acc[nt] = __builtin_amdgcn_wmma_f32_16x16x4_f32(
              false, a, false, bf, (short)0, acc[nt], false, false);
        }
      }
    }
  }

  // fused epilogue + store
  float* Cb = C + (long)b * sC;
  long eb = (long)b * D2;
#pragma unroll
  for (int nt = 0; nt < 8; ++nt) {
#pragma unroll
    for (int r = 0; r < 8; ++r) {
      int m = m0 + r + lhalf * 8;
      int n = nt * 16 + l15;
      long off = (long)m * DD + n;
      float v = acc[nt][r];
      if (bias) v += bias[off];
      switch (mode) {
        case EP_SIGMOID: v = 1.f / (1.f + __expf(-v)); break;
        case EP_SIG_MUL: v = (1.f / (1.f + __expf(-v))) * E1[eb + off]; break;
        case EP_GRUMIX: {
          float u = E1[eb + off], p = E2[eb + off];
          v = (1.f - u) * p + u * tanhf(v);
        } break;
        case EP_NONE: break;
        case EP_RELU: v = fmaxf(v, 0.f); break;
        case EP_OUTMIX: v = 0.5f * (E1[eb + off] + fmaxf(v, 0.f)); break;
      }
      Cb[off] = v;
    }
  }
}

// ---------------------------------------------------------------------------
extern "C" void kernel_launch(void* const* d_in, const int* in_sizes, int n_in,
                              void* d_out, int out_size, void* d_ws, size_t ws_size,
                              hipStream_t stream) {
  const float* Ahat      = (const float*)d_in[0];
  const float* node_embs = (const float*)d_in[1];
  const float* ht        = (const float*)d_in[2];
  const float* prevD     = (const float*)d_in[3];
  const float* mapper_w  = (const float*)d_in[4];
  const float* mapper_b  = (const float*)d_in[5];
  const float* Wu = (const float*)d_in[6];
  const float* Uu = (const float*)d_in[7];
  const float* bu = (const float*)d_in[8];
  const float* Wr = (const float*)d_in[9];
  const float* Ur = (const float*)d_in[10];
  const float* br = (const float*)d_in[11];
  const float* Wh = (const float*)d_in[12];
  const float* Uh = (const float*)d_in[13];
  const float* bh = (const float*)d_in[14];
  const float* static_w = (const float*)d_in[15];

  // Output layout: out [B,D,D] | policy [B] | scorer [B,D] | entropy [B]
  float* out       = (float*)d_out;
  float* policy_o  = out + (long)BB * D2;          // 524288
  float* scorer_o  = policy_o + BB;                // 524320
  float* entropy_o = scorer_o + (long)BB * DD;     // 528416

  // Workspace layout (floats)
  float* ws      = (float*)d_ws;
  float* invnorm = ws;                              // 32
  float* scores  = ws + 32;                         // 64000
  float* di      = ws + 32 + (long)BB * NN;         // 4096
  int*   topk    = (int*)(di + (long)BB * DD);      // 2048 ints
  float* xb      = (float*)topk + (long)BB * KTOP;  // x, later t1
  float* neb     = xb  + (long)BB * D2;
  float* An      = neb + (long)BB * D2;
  float* upd     = An  + (long)BB * D2;
  float* rh      = upd + (long)BB * D2;             // rh, later t2
  float* Dw      = rh  + (long)BB * D2;
  float* h1      = Dw  + (long)BB * D2;
  float* t1 = xb;   // x dead after gate GEMMs
  float* t2 = rh;   // rh dead after Dw GEMM

  // Phase 1: scorer + scores
  k_scorer<<<BB, DD, 0, stream>>>(ht, mapper_w, mapper_b, scorer_o, invnorm);
  k_scores<<<dim3(NN / 8, BB), 256, 0, stream>>>(node_embs, scorer_o, invnorm, scores);

  // Phase 2: topk + entropy + policy
  k_topk<<<BB, 256, 0, stream>>>(scores, topk, entropy_o, policy_o);

  // Phase 3: gathers
  k_di<<<BB, DD, 0, stream>>>(Ahat, topk, di);
  k_An<<<BB, 256, 0, stream>>>(Ahat, topk, di, An);
  k_sel<<<BB, 256, 0, stream>>>(node_embs, scores, topk, xb, neb);

  // Phase 4: GRU gate GEMMs (dual-operand, fused epilogues)
  // update = sigmoid(Wu@x + Uu@prevD + bu)
  k_gemm128<<<BB, 256, 0, stream>>>(Wu, 0, xb, D2, Uu, 0, prevD, D2, bu,
                                    nullptr, nullptr, upd, D2, EP_SIGMOID);
  // rh = sigmoid(Wr@x + Ur@prevD + br) * prevD
  k_gemm128<<<BB, 256, 0, stream>>>(Wr, 0, xb, D2, Ur, 0, prevD, D2, br,
                                    prevD, nullptr, rh, D2, EP_SIG_MUL);
  // Dw = (1-update)*prevD + update*tanh(Wh@x + Uh@rh + bh)
  k_gemm128<<<BB, 256, 0, stream>>>(Wh, 0, xb, D2, Uh, 0, rh, D2, bh,
                                    upd, prevD, Dw, D2, EP_GRUMIX);

  // Phase 5: GCN GEMMs
  // t1 = ne @ Dw
  k_gemm128<<<BB, 256, 0, stream>>>(neb, D2, Dw, D2, nullptr, 0, nullptr, 0,
                                    nullptr, nullptr, nullptr, t1, D2, EP_NONE);
  // h1 = relu(An @ t1)
  k_gemm128<<<BB, 256, 0, stream>>>(An, D2, t1, D2, nullptr, 0, nullptr, 0,
                                    nullptr, nullptr, nullptr, h1, D2, EP_RELU);
  // t2 = h1 @ static_w
  k_gemm128<<<BB, 256, 0, stream>>>(h1, D2, static_w, 0, nullptr, 0, nullptr, 0,
                                    nullptr, nullptr, nullptr, t2, D2, EP_NONE);
  // out = 0.5*(h1 + relu(An @ t2))
  k_gemm128<<<BB, 256, 0, stream>>>(An, D2, t2, D2, nullptr, 0, nullptr, 0,
                                    nullptr, h1, nullptr, out, D2, EP_OUTMIX);

  (void)in_sizes; (void)n_in; (void)out_size; (void)ws_size;
}